// MultiStagePooling_66537633349679
// MI455X (gfx1250) — compile-verified
//
#include <hip/hip_runtime.h>
#include <math.h>

typedef float v2f __attribute__((ext_vector_type(2)));
typedef float v8f __attribute__((ext_vector_type(8)));

#define HIDDEN 256
#define SCAP   2048   // LDS score-buffer capacity (max segment ~100 in this workload)

__device__ __forceinline__ int lower_bound(const int* __restrict__ arr, int n, int val) {
    int lo = 0, hi = n;
    while (lo < hi) {
        int mid = (lo + hi) >> 1;
        if (arr[mid] < val) lo = mid + 1; else hi = mid;
    }
    return lo;
}

__global__ __launch_bounds__(256)
void multi_stage_pool_kernel(const float* __restrict__ h,
                             const float* __restrict__ node_mult,
                             const float* __restrict__ pressure,
                             const float* __restrict__ attn_weight,
                             const float* __restrict__ attn_bias,
                             const float* __restrict__ W_pressure,
                             const int*   __restrict__ batch,
                             float* __restrict__ out,
                             float* __restrict__ ws_scores,
                             int N, int G)
{
    const int g   = blockIdx.x;
    const int tid = threadIdx.x;

    __shared__ int   sh_se[2];
    __shared__ float s_w[HIDDEN];
    __shared__ float s_red[256];
    __shared__ float s_scores[SCAP];   // scores, then reused as phase-B reduce scratch
    __shared__ float sh_smax, sh_denom, sh_tmult;

    // Preload attention weights to LDS (WMMA B operand source)
    s_w[tid] = attn_weight[tid];

    if (tid == 0) {
        sh_se[0] = lower_bound(batch, N, g);
        sh_se[1] = lower_bound(batch, N, g + 1);
    }
    __syncthreads();
    const int s = sh_se[0];
    const int e = sh_se[1];
    const int n = e - s;

    const float p_norm = pressure[g] * (1.0f / 300.0f);
    const size_t obase = (size_t)g * (2 * HIDDEN);

    if (n == 0) {
        out[obase + tid]          = p_norm * W_pressure[tid];
        out[obase + HIDDEN + tid] = -INFINITY;
        return;
    }

    float* scbuf;
    int    scoff;
    if (n <= SCAP) { scbuf = s_scores;  scoff = s; }
    else           { scbuf = ws_scores; scoff = 0; }

    const float bias = attn_bias[0];

    // ---------- Phase A: raw scores via V_WMMA_F32_16X16X4_F32 ----------
    {
        const int wave = tid >> 5;
        const int lane = tid & 31;
        const int m    = lane & 15;   // row within 16-node tile
        const int hi   = lane >> 4;   // K half select (K=0,1 vs K=2,3)
        const int ntiles = (n + 15) >> 4;

        for (int t = wave; t < ntiles; t += 8) {
            const int rowbase = s + (t << 4);
            // Clamp out-of-segment rows: always-in-bounds load; garbage C rows
            // are never extracted (guarded by r < e below). No exec masking.
            const int rowc = min(rowbase + m, e - 1);
            const float* hrow = h + (size_t)rowc * HIDDEN;

            v8f c0 = {0.f,0.f,0.f,0.f,0.f,0.f,0.f,0.f};
            v8f c1 = {0.f,0.f,0.f,0.f,0.f,0.f,0.f,0.f};
            #pragma unroll 4
            for (int kk = 0; kk < 64; kk += 2) {
                const int k0 = (kk << 2) + (hi << 1);
                const v2f a0 = *(const v2f*)(hrow + k0);
                const v2f b0 = *(const v2f*)(&s_w[k0]);
                c0 = __builtin_amdgcn_wmma_f32_16x16x4_f32(
                        false, a0, false, b0, (short)0, c0, false, false);
                const int k1 = k0 + 4;
                const v2f a1 = *(const v2f*)(hrow + k1);
                const v2f b1 = *(const v2f*)(&s_w[k1]);
                c1 = __builtin_amdgcn_wmma_f32_16x16x4_f32(
                        false, a1, false, b1, (short)0, c1, false, false);
            }
            const v8f c = c0 + c1;

            // Extract column 0: lane 0 holds rows rowbase+0..7 (VGPR v -> M=v),
            // lane 16 holds rows rowbase+8..15 (VGPR v -> M=v+8).
            if (m == 0) {
                #pragma unroll
                for (int v = 0; v < 8; ++v) {
                    const int r = rowbase + (hi << 3) + v;
                    if (r < e) {
                        const float mu = node_mult[r];
                        scbuf[r - scoff] = c[v] + bias + logf(fmaxf(mu, 1.0f));
                    }
                }
            }
        }
    }
    __threadfence();   // scbuf may be global in the fallback path
    __syncthreads();

    // ---------- Phase A2: segment max of raw scores ----------
    {
        float lm = -INFINITY;
        for (int i = s + tid; i < e; i += 256) lm = fmaxf(lm, scbuf[i - scoff]);
        s_red[tid] = lm;
        __syncthreads();
        #pragma unroll
        for (int off = 128; off > 0; off >>= 1) {
            if (tid < off) s_red[tid] = fmaxf(s_red[tid], s_red[tid + off]);
            __syncthreads();
        }
        if (tid == 0) sh_smax = s_red[0];
        __syncthreads();
    }
    const float smax = sh_smax;

    // ---------- Phase A3: exp(score - smax) in place; reduce denom & total_mult ----------
    {
        float ld = 0.f, lmlt = 0.f;
        for (int i = s + tid; i < e; i += 256) {
            const float ei = expf(scbuf[i - scoff] - smax);
            scbuf[i - scoff] = ei;
            ld   += ei;
            lmlt += node_mult[i];
        }
        s_red[tid] = ld;
        __syncthreads();
        #pragma unroll
        for (int off = 128; off > 0; off >>= 1) {
            if (tid < off) s_red[tid] += s_red[tid + off];
            __syncthreads();
        }
        if (tid == 0) sh_denom = s_red[0];
        __syncthreads();

        s_red[tid] = lmlt;
        __syncthreads();
        #pragma unroll
        for (int off = 128; off > 0; off >>= 1) {
            if (tid < off) s_red[tid] += s_red[tid + off];
            __syncthreads();
        }
        if (tid == 0) sh_tmult = s_red[0];
    }
    __threadfence();
    __syncthreads();

    // ---------- Phase B: stream segment rows with float4 (global_load_b128) ----------
    // 4 row-phases x 64 column-groups; thread = (p, cg) streams columns 4*cg..4*cg+3
    // over rows s+p, s+p+4, ...  Cross-phase reduction via LDS afterwards.
    const int p  = tid >> 6;        // row phase 0..3
    const int cg = tid & 63;        // column group 0..63

    float4 mean4 = make_float4(0.f, 0.f, 0.f, 0.f);
    float4 attn4 = make_float4(0.f, 0.f, 0.f, 0.f);
    float4 max4  = make_float4(-INFINITY, -INFINITY, -INFINITY, -INFINITY);

    for (int i = s + p; i < e; i += 4) {
        const float4 hv = *(const float4*)(h + (size_t)i * HIDDEN + (cg << 2));
        const float  ei = scbuf[i - scoff];   // broadcast within phase
        const float  mi = node_mult[i];       // broadcast within phase
        mean4.x += hv.x * mi;  mean4.y += hv.y * mi;
        mean4.z += hv.z * mi;  mean4.w += hv.w * mi;
        attn4.x += hv.x * ei;  attn4.y += hv.y * ei;
        attn4.z += hv.z * ei;  attn4.w += hv.w * ei;
        max4.x = fmaxf(max4.x, hv.x);  max4.y = fmaxf(max4.y, hv.y);
        max4.z = fmaxf(max4.z, hv.z);  max4.w = fmaxf(max4.w, hv.w);
    }

    // exp values fully consumed; safe to reuse s_scores as reduce scratch
    __syncthreads();

    // Thread (p,cg) stores its float4 at s_scores[tid*4 .. +3] == s_scores[p*256 + column]
    // so the gather for output column `tid` is simply s_scores[p*256 + tid]. Conflict-free.
    float meansum, attnsum, maxv;

    *(float4*)(&s_scores[tid << 2]) = mean4;
    __syncthreads();
    meansum = s_scores[tid] + s_scores[256 + tid] + s_scores[512 + tid] + s_scores[768 + tid];
    __syncthreads();

    *(float4*)(&s_scores[tid << 2]) = attn4;
    __syncthreads();
    attnsum = s_scores[tid] + s_scores[256 + tid] + s_scores[512 + tid] + s_scores[768 + tid];
    __syncthreads();

    *(float4*)(&s_scores[tid << 2]) = max4;
    __syncthreads();
    maxv = fmaxf(fmaxf(s_scores[tid], s_scores[256 + tid]),
                 fmaxf(s_scores[512 + tid], s_scores[768 + tid]));

    const float tm   = fmaxf(sh_tmult, 1e-8f);
    const float mean = meansum / tm;
    const float attn = attnsum / sh_denom;             // denom >= 1 (contains exp(0))
    const float comb = 0.5f * (mean + attn) + p_norm * W_pressure[tid];

    out[obase + tid]          = comb;
    out[obase + HIDDEN + tid] = maxv;
}

extern "C" void kernel_launch(void* const* d_in, const int* in_sizes, int n_in,
                              void* d_out, int out_size, void* d_ws, size_t ws_size,
                              hipStream_t stream) {
    (void)n_in; (void)out_size; (void)ws_size;
    const float* h           = (const float*)d_in[0];
    const float* node_mult   = (const float*)d_in[1];
    const float* pressure    = (const float*)d_in[2];
    const float* attn_weight = (const float*)d_in[3];
    const float* attn_bias   = (const float*)d_in[4];
    const float* W_pressure  = (const float*)d_in[5];
    const int*   batch       = (const int*)d_in[6];

    const int N = in_sizes[1];   // node count (node_mult length)
    const int G = in_sizes[2];   // graph count (pressure length)

    float* out       = (float*)d_out;
    float* ws_scores = (float*)d_ws;   // N floats fallback score buffer

    multi_stage_pool_kernel<<<G, 256, 0, stream>>>(
        h, node_mult, pressure, attn_weight, attn_bias, W_pressure, batch,
        out, ws_scores, N, G);
}